// Nms_83958020702341
// MI455X (gfx1250) — compile-verified
//
#include <hip/hip_runtime.h>
#include <stdint.h>
#include <stddef.h>

// ---------------------------------------------------------------------------
// NMS for N=16384 boxes, THRESHOLD=0.5, MAX_SIZE=1000, output int32[1000].
// Pipeline: build 64-bit sort keys -> bitonic sort (descending score, then
// ascending original index) -> gather sorted boxes/areas -> pairwise
// suppression-mask matrix (WMMA f32 16x16x4 computes area_i+area_j tiles,
// ballot_w32 packs bits) -> single-workgroup sequential greedy reduce.
// Workspace: ~32.5 MB (mask matrix 16384 x 512 u32 = 32 MB, L2-resident).
// ---------------------------------------------------------------------------

#define N_BOX    16384
#define WORDS    (N_BOX / 32)      // 512 mask words per row
#define NMS_THR  0.5f
#define MAX_KEEP 1000

typedef __attribute__((ext_vector_type(2))) float v2f;
typedef __attribute__((ext_vector_type(8))) float v8f;
typedef int tdm_v4i __attribute__((vector_size(16)));   // matches builtin's V4i

#if __has_builtin(__builtin_amdgcn_wmma_f32_16x16x4_f32)
#define HAVE_WMMA4 1
#endif

#if __has_builtin(__builtin_amdgcn_global_load_async_to_lds_b128) && \
    __has_builtin(__builtin_amdgcn_s_wait_asynccnt)
#define HAVE_ASYNC_LDS 1
#endif

__device__ __forceinline__ uint32_t ballot32(bool p) {
#if __has_builtin(__builtin_amdgcn_ballot_w32)
  return __builtin_amdgcn_ballot_w32(p);
#else
  return (uint32_t)__ballot(p ? 1 : 0);
#endif
}

// --- 1. sort keys: descending score, ties -> ascending original index -------
// score in [0,1) is non-negative float => IEEE bits are order-preserving.
// key = (score_bits << 32) | ~index   (larger key first = higher score,
// and for equal scores smaller index gives larger ~index).
__global__ void build_keys_kernel(const float* __restrict__ scores,
                                  uint64_t* __restrict__ keys) {
  int i = blockIdx.x * blockDim.x + threadIdx.x;
  if (i < N_BOX) {
    uint32_t sb = __float_as_uint(scores[i]);
    keys[i] = ((uint64_t)sb << 32) | (uint32_t)(~(uint32_t)i);
  }
}

// --- 2. bitonic sort step (global memory, descending) -----------------------
__global__ void bitonic_step_kernel(uint64_t* __restrict__ keys, int k, int j) {
  int i = blockIdx.x * blockDim.x + threadIdx.x;
  int ixj = i ^ j;
  if (ixj > i && ixj < N_BOX) {
    uint64_t a = keys[i];
    uint64_t b = keys[ixj];
    bool desc = ((i & k) == 0);          // overall descending order
    if (desc ? (a < b) : (a > b)) {
      keys[i] = b;
      keys[ixj] = a;
    }
  }
}

// --- 3. gather sorted boxes, areas, original indices ------------------------
__global__ void gather_kernel(const uint64_t* __restrict__ keys,
                              const float4* __restrict__ rois,
                              float4* __restrict__ boxes,
                              float* __restrict__ area,
                              int* __restrict__ order) {
  int p = blockIdx.x * blockDim.x + threadIdx.x;
  if (p < N_BOX) {
    uint32_t idx = ~(uint32_t)(keys[p] & 0xFFFFFFFFull);
    float4 b = rois[idx];
    boxes[p] = b;
    area[p] = (b.z - b.x) * (b.w - b.y);
    order[p] = (int)idx;
  }
}

// --- 4. pairwise suppression-mask matrix ------------------------------------
// One wave32 handles a 16-row x 32-col tile => one u32 word per row.
// Two 16x16 sub-tiles; area_i + area_j via V_WMMA_F32_16X16X4_F32:
//   A (16x4): col K0 = area_row, K1 = 1.0, K2/K3 = 0
//   B (4x16): row K0 = 1.0, K1 = area_col, K2/K3 = 0
// D[m][n] = area_row[m]*1 + 1*area_col[n]  (bit-exact f32 sum, RNE).
// D layout (32-bit C/D 16x16): VGPR r, lanes 0-15 -> M=r,N=lane;
// lanes 16-31 -> M=r+8,N=lane-16.  A layout: lanes 0-15 hold K=0,1 in the two
// VGPRs (lanes 16-31 hold K=2,3 == zeros).  B analogous.
__global__ void __launch_bounds__(256)
iou_mask_kernel(const float4* __restrict__ boxes,
                const float* __restrict__ area,
                uint32_t* __restrict__ mask) {
  __shared__ float4 sRow[8][16];   // per-wave row-tile boxes
  __shared__ float4 sCol[8][32];   // per-wave col-tile boxes

  const int lane = threadIdx.x & 31;
  const int wv   = threadIdx.x >> 5;
  const uint32_t wid = blockIdx.x * 8u + (uint32_t)wv;  // 0 .. 1024*512-1
  const int rt = (int)(wid >> 9);          // row tile index (16 rows each)
  const int cw = (int)(wid & 511u);        // mask-word (32-col) index
  const int rowBase = rt * 16;
  const int colBase = cw * 32;

  if (lane < 16) sRow[wv][lane] = boxes[rowBase + lane];

#if defined(HAVE_ASYNC_LDS)
  // CDNA5 async global->LDS copy, tracked by ASYNCcnt.
  __builtin_amdgcn_global_load_async_to_lds_b128(
      (__attribute__((address_space(1))) tdm_v4i*)(boxes + colBase + lane),
      (__attribute__((address_space(3))) tdm_v4i*)(&sCol[wv][lane]),
      0, 0);
  __builtin_amdgcn_s_wait_asynccnt(0);
#else
  sCol[wv][lane] = boxes[colBase + lane];
#endif
  __syncthreads();

  // A operand (shared by both sub-tiles)
  float arow = area[rowBase + (lane & 15)];
  v2f a;
  a.x = (lane < 16) ? arow : 0.0f;
  a.y = (lane < 16) ? 1.0f : 0.0f;

  const int  mloc_off = (lane >= 16) ? 8 : 0;
  const int  nl       = lane & 15;

  uint32_t bal0[8], bal1[8];

  #pragma unroll
  for (int s = 0; s < 2; ++s) {
    const int c = colBase + s * 16;
    float acol = area[c + nl];
    v2f b;
    b.x = (lane < 16) ? 1.0f : 0.0f;
    b.y = (lane < 16) ? acol : 0.0f;

    v8f asum = {};
#if defined(HAVE_WMMA4)
    // emits v_wmma_f32_16x16x4_f32
    asum = __builtin_amdgcn_wmma_f32_16x16x4_f32(
        /*neg_a=*/false, a, /*neg_b=*/false, b,
        /*c_mod=*/(short)0, asum, /*reuse_a=*/false, /*reuse_b=*/false);
#endif

    const float4 cb = sCol[wv][s * 16 + nl];
    #pragma unroll
    for (int r = 0; r < 8; ++r) {
      const int mloc = r + mloc_off;
      const float4 rb = sRow[wv][mloc];
      float ix1 = fmaxf(rb.x, cb.x);
      float iy1 = fmaxf(rb.y, cb.y);
      float ix2 = fminf(rb.z, cb.z);
      float iy2 = fminf(rb.w, cb.w);
      float iw = fmaxf(ix2 - ix1, 0.0f);
      float ih = fmaxf(iy2 - iy1, 0.0f);
      float inter = iw * ih;
#if defined(HAVE_WMMA4)
      float sum_area = asum[r];
#else
      float sum_area = ((rb.z - rb.x) * (rb.w - rb.y)) +
                       ((cb.z - cb.x) * (cb.w - cb.y));
#endif
      float uni = sum_area - inter;
      float iou = inter / fmaxf(uni, 1e-10f);
      const int m = rowBase + mloc;
      const int n = c + nl;
      bool pred = (iou > NMS_THR) && (n > m);
      uint32_t bal = ballot32(pred);
      if (s == 0) bal0[r] = bal; else bal1[r] = bal;
    }
  }

  // Assemble one u32 word per row.  For sub-tile ballots:
  //  bits[15:0]  = cols c..c+15 of row rowBase+r
  //  bits[31:16] = cols c..c+15 of row rowBase+r+8
  uint32_t word = 0;
  #pragma unroll
  for (int r = 0; r < 8; ++r) {
    uint32_t lo0 = bal0[r], lo1 = bal1[r];
    uint32_t wlow  = (lo0 & 0xFFFFu) | ((lo1 & 0xFFFFu) << 16);
    uint32_t whigh = (lo0 >> 16)     | (lo1 & 0xFFFF0000u);
    if (lane == r)     word = wlow;
    if (lane == r + 8) word = whigh;
  }
  if (lane < 16) {
    mask[(size_t)(rowBase + lane) * WORDS + cw] = word;
  }
}

// --- 5. sequential greedy reduce (single workgroup) -------------------------
__global__ void __launch_bounds__(1024)
nms_reduce_kernel(const uint32_t* __restrict__ mask,
                  const int* __restrict__ order,
                  int* __restrict__ out) {
  __shared__ uint32_t rem[WORDS];   // removed-bit per sorted box
  __shared__ int s_cnt;
  __shared__ int s_alive;

  const int t = threadIdx.x;
  for (int w = t; w < WORDS; w += 1024) rem[w] = 0u;
  if (t == 0) s_cnt = 0;
  __syncthreads();

  for (int i = 0; i < N_BOX; ++i) {
    if (t == 0) {
      bool alive = ((rem[i >> 5] >> (i & 31)) & 1u) == 0u;
      s_alive = alive ? 1 : 0;
      if (alive) {
        if (s_cnt < MAX_KEEP) out[s_cnt] = order[i];
        s_cnt = s_cnt + 1;
      }
    }
    __syncthreads();
    if (s_alive) {
      const uint32_t* row = mask + (size_t)i * WORDS;
      if (t < WORDS) {
        rem[t] |= row[t];
        // speculative prefetch of a future row (-> global_prefetch_b8)
        __builtin_prefetch(row + 4 * WORDS + t, 0, 0);
      }
    }
    __syncthreads();
  }

  // pad remainder with -1
  int cnt = s_cnt;
  for (int p = t; p < MAX_KEEP; p += 1024) {
    if (p >= cnt) out[p] = -1;
  }
}

// ---------------------------------------------------------------------------
extern "C" void kernel_launch(void* const* d_in, const int* in_sizes, int n_in,
                              void* d_out, int out_size, void* d_ws, size_t ws_size,
                              hipStream_t stream) {
  (void)in_sizes; (void)n_in; (void)out_size; (void)ws_size;

  const float4* rois   = (const float4*)d_in[0];   // (16384, 4) f32
  const float*  scores = (const float*)d_in[1];    // (16384,)   f32
  int* out = (int*)d_out;                          // int32[1000]

  uint8_t* ws = (uint8_t*)d_ws;
  uint64_t* keys  = (uint64_t*)(ws + 0);                 // 128 KB
  float4*   boxes = (float4*)  (ws + 131072);            // 256 KB
  float*    area  = (float*)   (ws + 393216);            //  64 KB
  int*      order = (int*)     (ws + 458752);            //  64 KB
  uint32_t* mask  = (uint32_t*)(ws + 524288);            //  32 MB

  const int T = 256;
  const int B = N_BOX / T;   // 64

  build_keys_kernel<<<B, T, 0, stream>>>(scores, keys);

  for (int k = 2; k <= N_BOX; k <<= 1) {
    for (int j = k >> 1; j > 0; j >>= 1) {
      bitonic_step_kernel<<<B, T, 0, stream>>>(keys, k, j);
    }
  }

  gather_kernel<<<B, T, 0, stream>>>(keys, rois, boxes, area, order);

  // 1024 row-tiles * 512 col-words = 524288 waves; 8 waves per 256-thr block.
  iou_mask_kernel<<<65536, 256, 0, stream>>>(boxes, area, mask);

  nms_reduce_kernel<<<1, 1024, 0, stream>>>(mask, order, out);
}